// FastBEVProjection_11656541241444
// MI455X (gfx1250) — compile-verified
//
#include <hip/hip_runtime.h>
#include <cstdint>

#define BEV_S   256
#define BEV_HW  (BEV_S * BEV_S)
#define NCH     64
#define NBINS   10
#define PLANE_BYTES 262144            // BEV_HW * 4

// ---------------------------------------------------------------------------
// gfx1250 native global atomics, GVS addressing: uniform SGPR base + shared
// 32-bit byte-offset VGPR + constant plane offset folded into the signed
// 24-bit instruction offset field. Device scope, non-returning (STOREcnt,
// implicit wait at s_endpgm).
// ---------------------------------------------------------------------------
template <int IMM>
__device__ __forceinline__ void gadd_f32(uint64_t sbase, uint32_t voff, float v) {
    asm volatile("global_atomic_add_f32 %0, %1, %2 offset:%3 scope:SCOPE_DEV"
                 :: "v"(voff), "v"(v), "s"(sbase), "n"(IMM) : "memory");
}
template <int IMM>
__device__ __forceinline__ void gmax_f32(uint64_t sbase, uint32_t voff, float v) {
    asm volatile("global_atomic_max_num_f32 %0, %1, %2 offset:%3 scope:SCOPE_DEV"
                 :: "v"(voff), "v"(v), "s"(sbase), "n"(IMM) : "memory");
}
template <int IMM>
__device__ __forceinline__ void gadd_u32(uint64_t sbase, uint32_t voff, uint32_t v) {
    asm volatile("global_atomic_add_u32 %0, %1, %2 offset:%3 scope:SCOPE_DEV"
                 :: "v"(voff), "v"(v), "s"(sbase), "n"(IMM) : "memory");
}

// ---------------------------------------------------------------------------
// Kernel 1: initialize d_out. Channel 1 (max accumulator) -> -10.0f,
// everything else -> 0.0f (zero bits double as u32 counters).
// Re-runs every launch => graph-replay safe. b128 stores.
// ---------------------------------------------------------------------------
__global__ void bev_init(float4* __restrict__ out4, int total4) {
    int i = blockIdx.x * blockDim.x + threadIdx.x;
    if (i >= total4) return;
    int ch = ((i * 4) / BEV_HW) & (NCH - 1);
    float v = (ch == 1) ? -10.0f : 0.0f;
    out4[i] = make_float4(v, v, v, v);
}

// ---------------------------------------------------------------------------
// Kernel 2: scatter. blockIdx.y = batch (keeps the output base scalar and
// avoids per-thread division). One point per thread, float4 coalesced load.
//
// Masked (out-of-range) points all route to cell 0 with z=0,i=0: their sum
// contributions are exactly 0 (skipped); their density contribution is
// wave-aggregated (ballot+popc -> one u32 atomic) into plane 0; their max
// contribution is one max(.,0) per wave. In-range points skip the density
// atomic entirely: density is reconstructed in finalize as plane0 + sum(bins).
// ---------------------------------------------------------------------------
__global__ void bev_scatter(const float4* __restrict__ pts,
                            float* __restrict__ out,
                            int nPerBatch) {
    const int b = blockIdx.y;
    const int i = blockIdx.x * blockDim.x + threadIdx.x;
    if (i >= nPerBatch) return;

    float4 p = pts[(size_t)b * nPerBatch + i];
    float x = p.x, y = p.y, z = p.z, w = p.w;

    bool inr = (x >= -50.0f) && (x < 50.0f) &&
               (y >= -50.0f) && (y < 50.0f) &&
               (z >=  -3.0f) && (z <  5.0f);

    const uint64_t obase = (uint64_t)(out + (size_t)b * (NCH * BEV_HW));

    uint32_t mb = __builtin_amdgcn_ballot_w32(!inr);
    if (mb != 0u && (threadIdx.x & 31u) == 0u) {
        gadd_u32<0>(obase, 0u, (uint32_t)__popc(mb));     // masked count -> plane 0, cell 0
        gmax_f32<1 * PLANE_BYTES>(obase, 0u, 0.0f);       // masked max contribution
    }

    if (inr) {
        // exact match of jnp: truncating cast of IEEE division, then clip
        int xi = (int)((x + 50.0f) / 0.390625f);
        int yi = (int)((y + 50.0f) / 0.390625f);
        xi = xi < 0 ? 0 : (xi > BEV_S - 1 ? BEV_S - 1 : xi);
        yi = yi < 0 ? 0 : (yi > BEV_S - 1 ? BEV_S - 1 : yi);
        uint32_t voff = (uint32_t)((yi * BEV_S + xi) * 4);

        // bin boundaries == f32 rounding of double(Z0 + i*span/10)
        int bin = (z >= -2.2f) + (z >= -1.4f) + (z >= -0.6f) + (z >= 0.2f) +
                  (z >=  1.0f) + (z >=  1.8f) + (z >=  2.6f) + (z >= 3.4f) +
                  (z >=  4.2f);

        gmax_f32<1 * PLANE_BYTES>(obase, voff, z);        // max_h
        gadd_f32<2 * PLANE_BYTES>(obase, voff, z);        // sum z
        gadd_f32<3 * PLANE_BYTES>(obase, voff, z * z);    // sum z^2
        gadd_f32<4 * PLANE_BYTES>(obase, voff, w);        // sum intensity
        gadd_u32<5 * PLANE_BYTES>(obase, voff + (uint32_t)bin * PLANE_BYTES, 1u);
    }
}

// ---------------------------------------------------------------------------
// Kernel 3: finalize. One thread per (batch, cell). density = plane0 (masked
// count, cell 0 only) + sum of the 10 bin counts. Transforms the 15 live
// channels in place; channels 15..63 stay zero from init.
// ---------------------------------------------------------------------------
__global__ void bev_finalize(float* __restrict__ out, int totalCells) {
    int i = blockIdx.x * blockDim.x + threadIdx.x;
    if (i >= totalCells) return;
    int b    = i >> 16;            // BEV_HW == 65536
    int cell = i & 0xFFFF;
    float* ob = out + (size_t)b * (NCH * BEV_HW) + cell;

    uint32_t cnt = *reinterpret_cast<uint32_t*>(ob);   // masked count (cell 0) or 0
    uint32_t bins[NBINS];
#pragma unroll
    for (int k = 0; k < NBINS; ++k) {
        bins[k] = reinterpret_cast<uint32_t*>(ob)[(5 + k) * BEV_HW];
        cnt += bins[k];
    }

    float density = (float)cnt;
    float m  = ob[1 * BEV_HW];     // -10 if untouched == where(density>0,...,-10)
    float sz = ob[2 * BEV_HW];
    float sq = ob[3 * BEV_HW];
    float si = ob[4 * BEV_HW];

    float denom = density > 1.0f ? density : 1.0f;
    float mean  = sz / denom;
    float var   = sq / denom - mean * mean;
    var = var > 0.0f ? var : 0.0f;
    float mx = m < -10.0f ? -10.0f : (m > 10.0f ? 10.0f : m);

    ob[0]          = log1pf(density);
    ob[1 * BEV_HW] = mx;
    ob[2 * BEV_HW] = mean;
    ob[3 * BEV_HW] = sqrtf(var);
    ob[4 * BEV_HW] = si / denom;
#pragma unroll
    for (int k = 0; k < NBINS; ++k)
        ob[(5 + k) * BEV_HW] = (float)bins[k];
}

// ---------------------------------------------------------------------------
extern "C" void kernel_launch(void* const* d_in, const int* in_sizes, int n_in,
                              void* d_out, int out_size, void* d_ws, size_t ws_size,
                              hipStream_t stream) {
    (void)n_in; (void)d_ws; (void)ws_size;
    const float4* pts = (const float4*)d_in[0];
    float* out = (float*)d_out;

    const int B      = 8;
    const int total  = in_sizes[0] / 4;      // 4,000,000 points
    const int nPerB  = total / B;            // 500,000
    const int total4 = out_size / 4;         // b128 init stores
    const int cells  = B * BEV_HW;

    hipLaunchKernelGGL(bev_init, dim3((total4 + 255) / 256), dim3(256), 0, stream,
                       (float4*)out, total4);
    hipLaunchKernelGGL(bev_scatter, dim3((nPerB + 255) / 256, B), dim3(256), 0, stream,
                       pts, out, nPerB);
    hipLaunchKernelGGL(bev_finalize, dim3((cells + 255) / 256), dim3(256), 0, stream,
                       out, cells);
}